// CMPGNN_8409545965935
// MI455X (gfx1250) — compile-verified
//
#include <hip/hip_runtime.h>
#include <math.h>

#define N_NODES 50000
#define E_EDGES 600000
#define F_INPUT 500
#define H 128
#define C_CLS 40
#define K_LAYERS 4

typedef float v2f __attribute__((ext_vector_type(2)));
typedef float v8f __attribute__((ext_vector_type(8)));

// Full-precision WMMA: D(16x16 f32) = A(16x4 f32) x B(4x16 f32) + C
// 8-arg form: (neg_a, A, neg_b, B, c_mod, C, reuse_a, reuse_b)
__device__ __forceinline__ v8f wmma_f32(v2f a, v2f b, v8f c) {
  return __builtin_amdgcn_wmma_f32_16x16x4_f32(false, a, false, b, (short)0, c,
                                               false, false);
}

// ---------------------------------------------------------------------------
// Q = x @ W_in^T + b_in        x:[N,500]  W_in:[128,500]  Q:[N,128]
// One wave computes one 16x16 output tile. 8 waves / block of 256.
// Tiles: (N/16)=3125 row-tiles x (H/16)=8 col-tiles = 25000 waves = 3125 blocks.
// ---------------------------------------------------------------------------
__global__ __launch_bounds__(256) void gemm_in_kernel(
    const float* __restrict__ x, const float* __restrict__ Win,
    const float* __restrict__ bin, float* __restrict__ Q) {
  const int wid  = blockIdx.x * 8 + (threadIdx.x >> 5);
  const int lane = threadIdx.x & 31;
  const int mt = wid >> 3;
  const int nt = wid & 7;
  const int m0 = mt * 16, n0 = nt * 16;
  const int half = lane >> 4;      // 0: K={0,1}, 1: K={2,3}
  const int l15  = lane & 15;      // M index for A lanes, N index for B lanes

  const float* arow = x   + (size_t)(m0 + l15) * F_INPUT + half * 2;
  const float* brow = Win + (size_t)(n0 + l15) * F_INPUT + half * 2;

  v8f acc = {0.f, 0.f, 0.f, 0.f, 0.f, 0.f, 0.f, 0.f};
#pragma unroll 5
  for (int k = 0; k < F_INPUT; k += 4) {
    v2f a = *(const v2f*)(arow + k);   // A[l15][k+2h .. +1]
    v2f b = *(const v2f*)(brow + k);   // B[k+2h..][n0+l15] = Win[n0+l15][k..]
    acc = wmma_f32(a, b, acc);
  }
  const float bv = bin[n0 + l15];
#pragma unroll
  for (int r = 0; r < 8; ++r) {
    Q[(size_t)(m0 + r + half * 8) * H + n0 + l15] = acc[r] + bv;
  }
}

// ---------------------------------------------------------------------------
// Fused per-layer GEMMs sharing the Q tile:
//   h3 = Q@W1^T ; h4 = Q@W2^T ; agg = relu(Q@Wf^T)   (agg seeds the scatter)
// ---------------------------------------------------------------------------
__global__ __launch_bounds__(256) void gemm3_kernel(
    const float* __restrict__ Q, const float* __restrict__ W1,
    const float* __restrict__ W2, const float* __restrict__ Wf,
    float* __restrict__ h3, float* __restrict__ h4, float* __restrict__ agg) {
  const int wid  = blockIdx.x * 8 + (threadIdx.x >> 5);
  const int lane = threadIdx.x & 31;
  const int mt = wid >> 3;
  const int nt = wid & 7;
  const int m0 = mt * 16, n0 = nt * 16;
  const int half = lane >> 4;
  const int l15  = lane & 15;

  const float* arow = Q  + (size_t)(m0 + l15) * H + half * 2;
  const float* b1r  = W1 + (size_t)(n0 + l15) * H + half * 2;
  const float* b2r  = W2 + (size_t)(n0 + l15) * H + half * 2;
  const float* bfr  = Wf + (size_t)(n0 + l15) * H + half * 2;

  v8f acc3 = {0.f, 0.f, 0.f, 0.f, 0.f, 0.f, 0.f, 0.f};
  v8f acc4 = acc3;
  v8f accf = acc3;
#pragma unroll 8
  for (int k = 0; k < H; k += 4) {
    v2f a  = *(const v2f*)(arow + k);
    v2f b1 = *(const v2f*)(b1r + k);
    v2f b2 = *(const v2f*)(b2r + k);
    v2f bf = *(const v2f*)(bfr + k);
    acc3 = wmma_f32(a, b1, acc3);
    acc4 = wmma_f32(a, b2, acc4);
    accf = wmma_f32(a, bf, accf);
  }
#pragma unroll
  for (int r = 0; r < 8; ++r) {
    const size_t o = (size_t)(m0 + r + half * 8) * H + n0 + l15;
    h3[o]  = acc3[r];
    h4[o]  = acc4[r];
    agg[o] = fmaxf(accf[r], 0.f);   // relu(Q@Wf^T) seeds the aggregate
  }
}

// ---------------------------------------------------------------------------
// Per edge e: s = sigmoid(-<h3[row], h4[col]>)
//             agg[col] += s*h3[row] - (1-s)*h4[row]
// One wave per edge; lane covers 4 of 128 features. h3/h4/agg are L2-resident.
// ---------------------------------------------------------------------------
__global__ __launch_bounds__(256) void edge_kernel(
    const int* __restrict__ row, const int* __restrict__ col,
    const float* __restrict__ h3, const float* __restrict__ h4,
    float* __restrict__ agg) {
  const int e = blockIdx.x * 8 + (threadIdx.x >> 5);
  if (e >= E_EDGES) return;
  const int lane = threadIdx.x & 31;
  const int r = row[e];
  const int c = col[e];

  const float4 a3 = *(const float4*)(h3 + (size_t)r * H + lane * 4);
  const float4 b4 = *(const float4*)(h4 + (size_t)c * H + lane * 4);
  float d = a3.x * b4.x + a3.y * b4.y + a3.z * b4.z + a3.w * b4.w;
#pragma unroll
  for (int off = 16; off >= 1; off >>= 1) d += __shfl_xor(d, off, 32);

  const float s   = 1.0f / (1.0f + expf(d));   // sigmoid(-d)
  const float sm1 = s - 1.0f;                  // -(1-s)

  const float4 a4 = *(const float4*)(h4 + (size_t)r * H + lane * 4);
  float* dst = agg + (size_t)c * H + lane * 4;
  atomicAdd(dst + 0, s * a3.x + sm1 * a4.x);
  atomicAdd(dst + 1, s * a3.y + sm1 * a4.y);
  atomicAdd(dst + 2, s * a3.z + sm1 * a4.z);
  atomicAdd(dst + 3, s * a3.w + sm1 * a4.w);
}

// ---------------------------------------------------------------------------
// Q = agg / max(||agg||_2, 1e-12)   one wave per node
// ---------------------------------------------------------------------------
__global__ __launch_bounds__(256) void node_norm_kernel(
    const float* __restrict__ agg, float* __restrict__ Q) {
  const int nd = blockIdx.x * 8 + (threadIdx.x >> 5);
  if (nd >= N_NODES) return;
  const int lane = threadIdx.x & 31;

  const float4 v = *(const float4*)(agg + (size_t)nd * H + lane * 4);
  float ss = v.x * v.x + v.y * v.y + v.z * v.z + v.w * v.w;
#pragma unroll
  for (int off = 16; off >= 1; off >>= 1) ss += __shfl_xor(ss, off, 32);

  const float inv = 1.0f / fmaxf(sqrtf(ss), 1e-12f);
  float4 o;
  o.x = v.x * inv; o.y = v.y * inv; o.z = v.z * inv; o.w = v.w * inv;
  *(float4*)(Q + (size_t)nd * H + lane * 4) = o;
}

// ---------------------------------------------------------------------------
// out = log_softmax(Q @ W_out^T + b_out)   one 128-thread block per node
// ---------------------------------------------------------------------------
__global__ __launch_bounds__(128) void out_kernel(
    const float* __restrict__ Q, const float* __restrict__ Wout,
    const float* __restrict__ bout, float* __restrict__ out) {
  __shared__ float q[H];
  __shared__ float lg[C_CLS];
  __shared__ float red[2];
  const int node = blockIdx.x;
  const int t = threadIdx.x;

  q[t] = Q[(size_t)node * H + t];
  __syncthreads();
  if (t < C_CLS) {
    const float* w = Wout + (size_t)t * H;
    float acc = bout[t];
#pragma unroll 8
    for (int k = 0; k < H; ++k) acc = fmaf(q[k], w[k], acc);
    lg[t] = acc;
  }
  __syncthreads();
  if (t == 0) {
    float m = -3.402823466e+38f;
    for (int i = 0; i < C_CLS; ++i) m = fmaxf(m, lg[i]);
    float ssum = 0.f;
    for (int i = 0; i < C_CLS; ++i) ssum += expf(lg[i] - m);
    red[0] = m;
    red[1] = logf(ssum);
  }
  __syncthreads();
  if (t < C_CLS) out[(size_t)node * C_CLS + t] = lg[t] - red[0] - red[1];
}

// ---------------------------------------------------------------------------
extern "C" void kernel_launch(void* const* d_in, const int* in_sizes, int n_in,
                              void* d_out, int out_size, void* d_ws,
                              size_t ws_size, hipStream_t stream) {
  const float* x    = (const float*)d_in[0];
  const int*   ei   = (const int*)d_in[1];
  const float* Win  = (const float*)d_in[2];
  const float* bin  = (const float*)d_in[3];
  const float* W1   = (const float*)d_in[4];
  const float* W2   = (const float*)d_in[5];
  const float* Wf   = (const float*)d_in[6];
  const float* Wout = (const float*)d_in[7];
  const float* bout = (const float*)d_in[8];

  const int* row = ei;              // edge_index[0,:]
  const int* col = ei + E_EDGES;    // edge_index[1,:]

  const size_t NH = (size_t)N_NODES * H;
  float* Q   = (float*)d_ws;
  float* h3  = Q + NH;
  float* h4  = h3 + NH;
  float* agg = h4 + NH;

  const int gemm_blocks = (N_NODES / 16) * (H / 16) / 8;  // 3125
  const int edge_blocks = E_EDGES / 8;                    // 75000
  const int node_blocks = N_NODES / 8;                    // 6250

  gemm_in_kernel<<<gemm_blocks, 256, 0, stream>>>(x, Win, bin, Q);

  for (int i = 0; i < K_LAYERS; ++i) {
    const size_t off = (size_t)i * H * H;
    gemm3_kernel<<<gemm_blocks, 256, 0, stream>>>(Q, W1 + off, W2 + off,
                                                  Wf + off, h3, h4, agg);
    edge_kernel<<<edge_blocks, 256, 0, stream>>>(row, col, h3, h4, agg);
    node_norm_kernel<<<node_blocks, 256, 0, stream>>>(agg, Q);
  }

  out_kernel<<<N_NODES, 128, 0, stream>>>(Q, Wout, bout, (float*)d_out);
}